// WeightedDiceLoss_67259187855449
// MI455X (gfx1250) — compile-verified
//
#include <hip/hip_runtime.h>
#include <hip/hip_bf16.h>

typedef float v2f __attribute__((ext_vector_type(2)));
typedef float v4f __attribute__((ext_vector_type(4)));
typedef float v8f __attribute__((ext_vector_type(8)));

#define NB      4      // batches
#define NC      16     // channels
#define KBLK    64     // k-blocks per batch
#define WAVES   4      // waves per workgroup
#define KT      64     // k-elements staged per wave per step
#define LSTRIDE 68     // padded LDS row stride in floats (bank-conflict-free wmma reads)

// Workspace layout (floats): cross[4][16][16], psum[4][16], tsum[4][16], inter[4][16]
#define WS_CROSS 0
#define WS_PSUM  (NB * NC * NC)
#define WS_TSUM  (WS_PSUM + NB * NC)
#define WS_INTER (WS_TSUM + NB * NC)
#define WS_TOTAL (WS_INTER + NB * NC)

__global__ void dice_zero_ws(float* ws) {
    int i = blockIdx.x * blockDim.x + threadIdx.x;
    if (i < WS_TOTAL) ws[i] = 0.0f;
}

// Heavy streaming kernel: per-batch Gram matrix via v_wmma_f32_16x16x4_f32,
// plus psum/tsum/inter folded into the same single pass over the data.
__global__ __launch_bounds__(WAVES * 32)
void dice_cross_kernel(const float* __restrict__ P, const float* __restrict__ T,
                       float* __restrict__ ws, int N) {
    __shared__ float lds[WAVES][NC * LSTRIDE];

    const int b    = blockIdx.x / KBLK;
    const int kb   = blockIdx.x % KBLK;
    const int wave = threadIdx.x >> 5;
    const int lane = threadIdx.x & 31;
    const int h    = lane >> 4;   // half-wave selector (K pair / M offset)
    const int c    = lane & 15;   // channel / column index

    const int kPerBlock = N / KBLK;             // 4096 for N = 262144
    const int kPerWave  = kPerBlock / WAVES;    // 1024
    const int k0        = kb * kPerBlock + wave * kPerWave;

    const float* Pb = P + (size_t)b * NC * N;
    const float* Tb = T + (size_t)b * NC * N;
    float* tile = lds[wave];

    v8f acc = {0.f, 0.f, 0.f, 0.f, 0.f, 0.f, 0.f, 0.f};
    float psum[8], tsum[8], intr[8];
#pragma unroll
    for (int j = 0; j < 8; ++j) { psum[j] = 0.f; tsum[j] = 0.f; intr[j] = 0.f; }

    for (int ks = k0; ks < k0 + kPerWave; ks += KT) {
        // ---- Stage a 16 x KT tile of P into LDS with coalesced float4 loads.
        // Lane l covers row (2*it + h), cols [4c, 4c+3]. While staging,
        // accumulate psum/tsum/inter so predict/target are read exactly once.
#pragma unroll
        for (int it = 0; it < 8; ++it) {
            const int row = 2 * it + h;
            const int col = c * 4;
            const size_t g = (size_t)row * N + ks + col;
            v4f pv = *(const v4f*)(Pb + g);
            v4f tv = *(const v4f*)(Tb + g);
            psum[it] += pv.x + pv.y + pv.z + pv.w;
            tsum[it] += tv.x + tv.y + tv.z + tv.w;
            intr[it] += pv.x * tv.x + pv.y * tv.y + pv.z * tv.z + pv.w * tv.w;
            *(v4f*)(tile + row * LSTRIDE + col) = pv;
        }
        // Prefetch next stage's cachelines (emits global_prefetch_b8)
        if (ks + KT < k0 + kPerWave) {
            __builtin_prefetch(Pb + (size_t)h * N + ks + KT + c * 4, 0, 1);
            __builtin_prefetch(Tb + (size_t)h * N + ks + KT + c * 4, 0, 1);
        }

        // ---- Gram update: 16 WMMA steps of K=4. For f32 WMMA the A-matrix
        // (row per lane) and B-matrix (column per lane) register layouts are
        // identical when B == A^T, so feed the same v2f as both operands.
        // Padded stride 68 -> 32 lanes x 2 words hit 64 distinct LDS banks.
#pragma unroll
        for (int kk = 0; kk < 16; ++kk) {
            v2f ab = *(const v2f*)(tile + c * LSTRIDE + kk * 4 + 2 * h);
            acc = __builtin_amdgcn_wmma_f32_16x16x4_f32(
                false, ab, false, ab, (short)0, acc, false, false);
        }
    }

    // ---- Flush Gram partial: C/D layout -> VGPR i holds (M = i + 8*h, N = c)
    float* crossAcc = ws + WS_CROSS;
#pragma unroll
    for (int i = 0; i < 8; ++i) {
        const int M = i + 8 * h;
        atomicAdd(&crossAcc[(b * NC + M) * NC + c], acc[i]);
    }

    // ---- Reduce scalar partials across the 16 lanes of each half-wave.
    // Accumulator j in half h belongs to channel (2*j + h).
    float* psumAcc = ws + WS_PSUM;
    float* tsumAcc = ws + WS_TSUM;
    float* intrAcc = ws + WS_INTER;
#pragma unroll
    for (int j = 0; j < 8; ++j) {
        float ps = psum[j], ts = tsum[j], in = intr[j];
        for (int off = 1; off < 16; off <<= 1) {
            ps += __shfl_xor(ps, off, 32);
            ts += __shfl_xor(ts, off, 32);
            in += __shfl_xor(in, off, 32);
        }
        if (c == 0) {
            const int ch = 2 * j + h;
            atomicAdd(&psumAcc[b * NC + ch], ps);
            atomicAdd(&tsumAcc[b * NC + ch], ts);
            atomicAdd(&intrAcc[b * NC + ch], in);
        }
    }
}

// Tiny finalize: 3 output scalars from the 1216-float workspace.
__global__ void dice_finalize_kernel(const float* __restrict__ ws,
                                     const float* __restrict__ weights,
                                     float* __restrict__ out) {
    const float* crossAcc = ws + WS_CROSS;
    const float* psumAcc  = ws + WS_PSUM;
    const float* tsumAcc  = ws + WS_TSUM;
    const float* intrAcc  = ws + WS_INTER;

    __shared__ float sDice, sInter;
    if (threadIdx.x == 0) { sDice = 0.f; sInter = 0.f; }
    __syncthreads();

    float dl = 0.f;
    for (int i = threadIdx.x; i < NB * NC; i += blockDim.x) {
        const float m = weights[i] > 0.f ? 1.f : 0.f;
        const float dice = (2.f * intrAcc[i]) / (psumAcc[i] + tsumAcc[i]);
        dl += (1.f - dice) * 0.5f * m;   // DICE_PENALTY = 0.5, SMOOTH = 0
    }
    float il = 0.f;
    for (int idx = threadIdx.x; idx < NB * NC * NC; idx += blockDim.x) {
        const int b = idx >> 8, ci = (idx >> 4) & 15, di = idx & 15;
        if (ci < di) {   // strict upper triangle
            const float mi = weights[b * NC + ci] > 0.f ? 1.f : 0.f;
            const float mj = weights[b * NC + di] > 0.f ? 1.f : 0.f;
            const float pd = (2.f * crossAcc[idx]) /
                             (psumAcc[b * NC + ci] + psumAcc[b * NC + di]);
            il += 0.5f * pd * mi * mj;   // INTERSECTION_PENALTY = 0.5
        }
    }
    atomicAdd(&sDice, dl);
    atomicAdd(&sInter, il);
    __syncthreads();
    if (threadIdx.x == 0) {
        out[0] = sDice + sInter;
        out[1] = sDice;
        out[2] = sInter;
    }
}

extern "C" void kernel_launch(void* const* d_in, const int* in_sizes, int n_in,
                              void* d_out, int out_size, void* d_ws, size_t ws_size,
                              hipStream_t stream) {
    const float* predict = (const float*)d_in[0];
    const float* target  = (const float*)d_in[1];
    const float* weights = (const float*)d_in[2];
    float* out = (float*)d_out;
    float* ws  = (float*)d_ws;

    const int N = in_sizes[0] / (NB * NC);   // H*W = 262144

    dice_zero_ws<<<(WS_TOTAL + 255) / 256, 256, 0, stream>>>(ws);
    dice_cross_kernel<<<NB * KBLK, WAVES * 32, 0, stream>>>(predict, target, ws, N);
    dice_finalize_kernel<<<1, 256, 0, stream>>>(ws, weights, out);
}